// BasicDGCNN_36679020708002
// MI455X (gfx1250) — compile-verified
//
#include <hip/hip_runtime.h>

#define BB 8
#define NN 4096
#define KK 20
#define MAXVV 64
#define EPSF 1e-5f

typedef float v2f __attribute__((ext_vector_type(2)));
typedef float v8f __attribute__((ext_vector_type(8)));

__device__ __forceinline__ float leakyf(float t) { return t > 0.f ? t : 0.2f * t; }

// ---------------------------------------------------------------------------
// PCA alignment: one block per batch.
// ---------------------------------------------------------------------------
__global__ void pca_kernel(const float* __restrict__ x, float* __restrict__ out,
                           float* __restrict__ pts1, float* __restrict__ xx1,
                           int o_aligned, int o_rot, int o_centroid, int o_pca, int o_gf) {
  int b = blockIdx.x, t = threadIdx.x;
  __shared__ float stage[256 * 12];
  __shared__ float sh[16];
  __shared__ int cnts[6];

  // phase 1: sums for mean + second moments
  float s[9];
  for (int j = 0; j < 9; ++j) s[j] = 0.f;
  for (int n = t; n < NN; n += 256) {
    const float* p = x + ((size_t)b * NN + n) * 5;
    float px = p[0], py = p[1], pz = p[2];
    s[0] += px; s[1] += py; s[2] += pz;
    s[3] += px * px; s[4] += px * py; s[5] += px * pz;
    s[6] += py * py; s[7] += py * pz; s[8] += pz * pz;
  }
  for (int j = 0; j < 9; ++j) stage[t * 12 + j] = s[j];
  __syncthreads();
  if (t < 9) {
    float acc = 0.f;
    for (int r = 0; r < 256; ++r) acc += stage[r * 12 + t];
    stage[3072 + t] = acc;  // park sums
  }
  if (t < 6) cnts[t] = 0;
  __syncthreads();
  if (t == 0) {
    float inv_n = 1.f / NN;
    float mx = stage[3072 + 0] * inv_n, my = stage[3072 + 1] * inv_n, mz = stage[3072 + 2] * inv_n;
    float cxx = stage[3072 + 3] * inv_n - mx * mx, cxy = stage[3072 + 4] * inv_n - mx * my;
    float cxz = stage[3072 + 5] * inv_n - mx * mz, cyy = stage[3072 + 6] * inv_n - my * my;
    float cyz = stage[3072 + 7] * inv_n - my * mz, czz = stage[3072 + 8] * inv_n - mz * mz;
    // closed-form eigenvalues (descending)
    float p1 = cxy * cxy + cxz * cxz + cyz * cyz;
    float q = (cxx + cyy + czz) / 3.f;
    float p2 = (cxx - q) * (cxx - q) + (cyy - q) * (cyy - q) + (czz - q) * (czz - q) + 2.f * p1;
    float pp = sqrtf(fmaxf(p2 / 6.f, 1e-30f));
    float ib = 1.f / pp;
    float b00 = (cxx - q) * ib, b01 = cxy * ib, b02 = cxz * ib;
    float b11 = (cyy - q) * ib, b12 = cyz * ib, b22 = (czz - q) * ib;
    float detB = b00 * (b11 * b22 - b12 * b12) - b01 * (b01 * b22 - b12 * b02) +
                 b02 * (b01 * b12 - b11 * b02);
    float r = fminf(fmaxf(detB * 0.5f, -1.f), 1.f);
    float phi = acosf(r) / 3.f;
    float l0 = q + 2.f * pp * cosf(phi);
    float l2 = q + 2.f * pp * cosf(phi + 2.0943951023931953f);
    float l1 = 3.f * q - l0 - l2;
    float lam[3] = {l0, l1, l2};
    float V[3][3];
    for (int a = 0; a < 3; ++a) {
      float m00 = cxx - lam[a], m11 = cyy - lam[a], m22 = czz - lam[a];
      float r0[3] = {m00, cxy, cxz}, r1[3] = {cxy, m11, cyz}, r2[3] = {cxz, cyz, m22};
      float c01[3] = {r0[1] * r1[2] - r0[2] * r1[1], r0[2] * r1[0] - r0[0] * r1[2],
                      r0[0] * r1[1] - r0[1] * r1[0]};
      float c02[3] = {r0[1] * r2[2] - r0[2] * r2[1], r0[2] * r2[0] - r0[0] * r2[2],
                      r0[0] * r2[1] - r0[1] * r2[0]};
      float c12[3] = {r1[1] * r2[2] - r1[2] * r2[1], r1[2] * r2[0] - r1[0] * r2[2],
                      r1[0] * r2[1] - r1[1] * r2[0]};
      float n01 = c01[0] * c01[0] + c01[1] * c01[1] + c01[2] * c01[2];
      float n02 = c02[0] * c02[0] + c02[1] * c02[1] + c02[2] * c02[2];
      float n12 = c12[0] * c12[0] + c12[1] * c12[1] + c12[2] * c12[2];
      float* cb = c01; float nb = n01;
      if (n02 > nb) { cb = c02; nb = n02; }
      if (n12 > nb) { cb = c12; nb = n12; }
      float inv = (nb > 1e-30f) ? rsqrtf(nb) : 0.f;
      float vx = cb[0] * inv, vy = cb[1] * inv, vz = cb[2] * inv;
      if (inv == 0.f) { vx = (a == 0); vy = (a == 1); vz = (a == 2); }
      V[0][a] = vx; V[1][a] = vy; V[2][a] = vz;
    }
    float det = V[0][0] * (V[1][1] * V[2][2] - V[1][2] * V[2][1]) -
                V[0][1] * (V[1][0] * V[2][2] - V[1][2] * V[2][0]) +
                V[0][2] * (V[1][0] * V[2][1] - V[1][1] * V[2][0]);
    if (det < 0.f) { V[0][2] = -V[0][2]; V[1][2] = -V[1][2]; V[2][2] = -V[2][2]; }
    sh[0] = mx; sh[1] = my; sh[2] = mz;
    for (int c = 0; c < 3; ++c)
      for (int a = 0; a < 3; ++a) sh[3 + c * 3 + a] = V[c][a];
    sh[12] = lam[0]; sh[13] = lam[1]; sh[14] = lam[2];
  }
  __syncthreads();
  float mu0 = sh[0], mu1 = sh[1], mu2 = sh[2];
  float Vl[9];
  for (int j = 0; j < 9; ++j) Vl[j] = sh[3 + j];
  // phase 3: projection sign counts (int atomics: deterministic values)
  int cp[3] = {0, 0, 0}, cn[3] = {0, 0, 0};
  for (int n = t; n < NN; n += 256) {
    const float* p = x + ((size_t)b * NN + n) * 5;
    float c0 = p[0] - mu0, c1 = p[1] - mu1, c2 = p[2] - mu2;
    for (int a = 0; a < 3; ++a) {
      float pr = c0 * Vl[0 * 3 + a] + c1 * Vl[1 * 3 + a] + c2 * Vl[2 * 3 + a];
      if (pr > 0.f) cp[a]++; else if (pr < 0.f) cn[a]++;
    }
  }
  for (int a = 0; a < 3; ++a) { atomicAdd(&cnts[a], cp[a]); atomicAdd(&cnts[3 + a], cn[a]); }
  __syncthreads();
  if (t == 0) {
    for (int a = 0; a < 3; ++a) {
      float sgn = (cnts[3 + a] > cnts[a]) ? -1.f : 1.f;
      sh[3 + 0 * 3 + a] *= sgn; sh[3 + 1 * 3 + a] *= sgn; sh[3 + 2 * 3 + a] *= sgn;
    }
    for (int c = 0; c < 3; ++c)
      for (int a = 0; a < 3; ++a) out[o_rot + (b * 3 + c) * 3 + a] = sh[3 + c * 3 + a];
    out[o_centroid + b * 3 + 0] = mu0;
    out[o_centroid + b * 3 + 1] = mu1;
    out[o_centroid + b * 3 + 2] = mu2;
  }
  __syncthreads();
  for (int j = 0; j < 9; ++j) Vl[j] = sh[3 + j];
  // phase 5: aligned coordinates, padded points, norms, extents
  float mn[3] = {3.4e38f, 3.4e38f, 3.4e38f}, mxv[3] = {-3.4e38f, -3.4e38f, -3.4e38f};
  for (int n = t; n < NN; n += 256) {
    const float* p = x + ((size_t)b * NN + n) * 5;
    float c0 = p[0] - mu0, c1 = p[1] - mu1, c2 = p[2] - mu2;
    float a0 = c0 * Vl[0] + c1 * Vl[3] + c2 * Vl[6];
    float a1 = c0 * Vl[1] + c1 * Vl[4] + c2 * Vl[7];
    float a2 = c0 * Vl[2] + c1 * Vl[5] + c2 * Vl[8];
    float s3 = p[3], s4 = p[4];
    size_t ro = (size_t)b * NN + n;
    out[o_aligned + ro * 3 + 0] = a0;
    out[o_aligned + ro * 3 + 1] = a1;
    out[o_aligned + ro * 3 + 2] = a2;
    float* qq = pts1 + ro * 8;
    qq[0] = a0; qq[1] = a1; qq[2] = a2; qq[3] = s3; qq[4] = s4;
    qq[5] = 0.f; qq[6] = 0.f; qq[7] = 0.f;
    xx1[ro] = a0 * a0 + a1 * a1 + a2 * a2 + s3 * s3 + s4 * s4;
    mn[0] = fminf(mn[0], a0); mn[1] = fminf(mn[1], a1); mn[2] = fminf(mn[2], a2);
    mxv[0] = fmaxf(mxv[0], a0); mxv[1] = fmaxf(mxv[1], a1); mxv[2] = fmaxf(mxv[2], a2);
  }
  for (int a = 0; a < 3; ++a) { stage[t * 12 + a] = mn[a]; stage[t * 12 + 3 + a] = mxv[a]; }
  __syncthreads();
  if (t < 3) {
    float m1 = 3.4e38f, m2 = -3.4e38f;
    for (int r2 = 0; r2 < 256; ++r2) {
      m1 = fminf(m1, stage[r2 * 12 + t]);
      m2 = fmaxf(m2, stage[r2 * 12 + 3 + t]);
    }
    float ext = m2 - m1;
    float vs = sh[12] + sh[13] + sh[14] + 1e-8f;
    float eg = sh[12 + t] / vs;
    out[o_pca + b * 6 + t] = eg;
    out[o_pca + b * 6 + 3 + t] = ext;
    out[o_gf + b * 198 + 192 + t] = eg;
    out[o_gf + b * 198 + 195 + t] = ext;
  }
}

// ---------------------------------------------------------------------------
// Fused Gram-matrix (fp32 WMMA 16x16x4) + exact top-20 kNN.
// 256 threads = 8 waves; wave w owns rows [w*32, w*32+32) of a 256-row block.
// Top-20 lists live in per-thread LDS slices (only the running threshold is
// kept in a VGPR) so the A fragments stay register-resident with no spills.
// ---------------------------------------------------------------------------
template <int CIN>
__global__ void knn_wmma_kernel(const float* __restrict__ pts, const float* __restrict__ xx,
                                int* __restrict__ idx) {
  constexpr int KC = CIN / 4;
  __shared__ float g_lds[256 * 17];   // 16x16 Gram tiles, stride 17 (bank-conflict pad)
  __shared__ float bd_lds[256 * KK];  // per-thread sorted top-K distances
  __shared__ int bi_lds[256 * KK];    // per-thread top-K indices
  int t = threadIdx.x;
  int lane = t & 31;
  int w = t >> 5;
  int b = blockIdx.x >> 4;  // 16 blocks per batch (N/256)
  int row_base = (blockIdx.x & 15) * 256;
  const float* P = pts + (size_t)b * NN * CIN;
  int m = lane & 15, h = lane >> 4;

  // A fragments for this wave's two 16-row tiles (layout: K = 2*h + component)
  v2f A0[KC], A1[KC];
  int r0 = row_base + w * 32 + m;
  int r1 = r0 + 16;
#pragma unroll
  for (int kc = 0; kc < KC; ++kc) {
    int kb = kc * 4 + 2 * h;
    v2f a0 = {P[(size_t)r0 * CIN + kb], P[(size_t)r0 * CIN + kb + 1]};
    v2f a1 = {P[(size_t)r1 * CIN + kb], P[(size_t)r1 * CIN + kb + 1]};
    A0[kc] = a0; A1[kc] = a1;
  }

  float* myd = &bd_lds[t * KK];
  int* myi = &bi_lds[t * KK];
#pragma unroll
  for (int q = 0; q < KK; ++q) { myd[q] = 3.4e38f; myi[q] = 0; }
  float thresh = 3.4e38f;

  int myrow = row_base + t;
  float xxr = xx[(size_t)b * NN + myrow];

  for (int ct = 0; ct < NN / 16; ++ct) {
    int col_base = ct * 16;
    v2f Bf[KC];
#pragma unroll
    for (int kc = 0; kc < KC; ++kc) {
      int kb = kc * 4 + 2 * h;
      const float* pc = P + (size_t)(col_base + m) * CIN + kb;
      v2f bb = {pc[0], pc[1]};
      Bf[kc] = bb;
    }
    v8f c0 = {};
    v8f c1 = {};
#pragma unroll
    for (int kc = 0; kc < KC; ++kc) {
      c0 = __builtin_amdgcn_wmma_f32_16x16x4_f32(false, A0[kc], false, Bf[kc], (short)0, c0,
                                                 false, false);
      c1 = __builtin_amdgcn_wmma_f32_16x16x4_f32(false, A1[kc], false, Bf[kc], (short)0, c1,
                                                 false, false);
    }
    // C layout: VGPR r -> M = r + 8*h, N = lane&15
    int colp = lane & 15;
    int rb = w * 32 + 8 * h;
#pragma unroll
    for (int r = 0; r < 8; ++r) {
      g_lds[(rb + r) * 17 + colp] = c0[r];
      g_lds[(rb + 16 + r) * 17 + colp] = c1[r];
    }
    __syncthreads();
    // each thread owns block-row t: exact top-20 update (ties -> lower index)
    const float* xb = xx + (size_t)b * NN + col_base;
    for (int j = 0; j < 16; ++j) {
      float g = g_lds[t * 17 + j];
      float d2 = xxr + xb[j] - 2.f * g;
      if (d2 < thresh) {
        int q = KK - 2;
        while (q >= 0 && d2 < myd[q]) {
          myd[q + 1] = myd[q];
          myi[q + 1] = myi[q];
          --q;
        }
        myd[q + 1] = d2;
        myi[q + 1] = col_base + j;
        thresh = myd[KK - 1];
      }
    }
    __syncthreads();
  }
  int* op = idx + ((size_t)b * NN + myrow) * KK;
#pragma unroll
  for (int q = 0; q < KK; ++q) op[q] = myi[q];
}

// ---------------------------------------------------------------------------
// Edge-conv 1 (cin=5, cout=32): stats pass (deterministic block partials)
// ---------------------------------------------------------------------------
__global__ void ec1_stats_kernel(const float* __restrict__ pts1, const int* __restrict__ idx1,
                                 const float* __restrict__ w1, float* __restrict__ partial) {
  __shared__ float w1s[320];
  __shared__ float stage[256 * 32];
  int t = threadIdx.x;
  for (int i = t; i < 320; i += 256) w1s[i] = w1[i];
  __syncthreads();
  size_t g = (size_t)blockIdx.x * 256 + t;
  size_t b = g / NN;
  float c[5];
  const float* cp = pts1 + g * 8;
  for (int i = 0; i < 5; ++i) c[i] = cp[i];
  float sum[32], sq[32];
#pragma unroll
  for (int o = 0; o < 32; ++o) { sum[o] = 0.f; sq[o] = 0.f; }
  const int* ip = idx1 + g * KK;
  for (int k = 0; k < KK; ++k) {
    int j = ip[k];
    const float* np = pts1 + (b * NN + j) * 8;
    float e[10];
#pragma unroll
    for (int i = 0; i < 5; ++i) { e[i] = c[i]; e[5 + i] = np[i] - c[i]; }
#pragma unroll
    for (int o = 0; o < 32; ++o) {
      float hh = 0.f;
#pragma unroll
      for (int i = 0; i < 10; ++i) hh += w1s[o * 10 + i] * e[i];
      sum[o] += hh; sq[o] += hh * hh;
    }
  }
  for (int pass = 0; pass < 2; ++pass) {
#pragma unroll
    for (int o = 0; o < 32; ++o) stage[t * 32 + o] = pass ? sq[o] : sum[o];
    __syncthreads();
    if (t < 32) {
      float acc = 0.f;
      for (int r = 0; r < 256; ++r) acc += stage[r * 32 + t];
      partial[(size_t)blockIdx.x * 64 + pass * 32 + t] = acc;
    }
    __syncthreads();
  }
}

// partial layout: [block][2C] (C sums then C sumsq)
__global__ void bn_finalize_kernel(const float* __restrict__ partial, float* __restrict__ stats,
                                   int C, int nblocks, float inv_cnt) {
  __shared__ float S[256];
  int t = threadIdx.x;
  if (t < 2 * C) {
    float acc = 0.f;
    for (int r = 0; r < nblocks; ++r) acc += partial[(size_t)r * (2 * C) + t];
    S[t] = acc;
  }
  __syncthreads();
  if (t < C) {
    float mean = S[t] * inv_cnt;
    float var = S[C + t] * inv_cnt - mean * mean;
    stats[t * 2] = mean;
    stats[t * 2 + 1] = rsqrtf(var + EPSF);
  }
}

__global__ void ec1_final_kernel(const float* __restrict__ pts1, const int* __restrict__ idx1,
                                 const float* __restrict__ w1, const float* __restrict__ stats,
                                 const float* __restrict__ g1, const float* __restrict__ b1,
                                 float* __restrict__ f1cn, float* __restrict__ f1nc,
                                 float* __restrict__ xx2) {
  __shared__ float w1s[320];
  __shared__ float st[64];
  __shared__ float gb[64];
  int t = threadIdx.x;
  for (int i = t; i < 320; i += 256) w1s[i] = w1[i];
  if (t < 64) st[t] = stats[t];
  if (t < 32) { gb[t] = g1[t]; gb[32 + t] = b1[t]; }
  __syncthreads();
  size_t g = (size_t)blockIdx.x * 256 + t;
  size_t b = g / NN, n = g % NN;
  float c[5];
  const float* cp = pts1 + g * 8;
  for (int i = 0; i < 5; ++i) c[i] = cp[i];
  float mv[32];
#pragma unroll
  for (int o = 0; o < 32; ++o) mv[o] = -3.4e38f;
  const int* ip = idx1 + g * KK;
  for (int k = 0; k < KK; ++k) {
    int j = ip[k];
    const float* np = pts1 + (b * NN + j) * 8;
    float e[10];
#pragma unroll
    for (int i = 0; i < 5; ++i) { e[i] = c[i]; e[5 + i] = np[i] - c[i]; }
#pragma unroll
    for (int o = 0; o < 32; ++o) {
      float hh = 0.f;
#pragma unroll
      for (int i = 0; i < 10; ++i) hh += w1s[o * 10 + i] * e[i];
      float v = leakyf((hh - st[o * 2]) * st[o * 2 + 1] * gb[o] + gb[32 + o]);
      mv[o] = fmaxf(mv[o], v);
    }
  }
  float ssq = 0.f;
#pragma unroll
  for (int o = 0; o < 32; ++o) {
    f1cn[(b * 32 + o) * NN + n] = mv[o];
    f1nc[g * 32 + o] = mv[o];
    ssq += mv[o] * mv[o];
  }
  xx2[g] = ssq;
}

// ---------------------------------------------------------------------------
// Edge-conv 2 (cin=32, cout=64): thread = (row, o-chunk of 16)
// ---------------------------------------------------------------------------
__global__ void ec2_stats_kernel(const float* __restrict__ pts2, const int* __restrict__ idx2,
                                 const float* __restrict__ w2, float* __restrict__ partial) {
  __shared__ float w2s[64 * 64];
  __shared__ float stage[256 * 32];
  int t = threadIdx.x;
  for (int i = t; i < 4096; i += 256) w2s[i] = w2[i];
  __syncthreads();
  int rl = t >> 2, oc = t & 3;
  size_t g = (size_t)blockIdx.x * 64 + rl;
  size_t b = g / NN;
  float c[32];
  const float* cp = pts2 + g * 32;
#pragma unroll
  for (int i = 0; i < 32; ++i) c[i] = cp[i];
  float sum[16], sq[16];
#pragma unroll
  for (int o = 0; o < 16; ++o) { sum[o] = 0.f; sq[o] = 0.f; }
  const int* ip = idx2 + g * KK;
  int obase = oc * 16;
  for (int k = 0; k < KK; ++k) {
    int j = ip[k];
    const float* np = pts2 + (b * NN + j) * 32;
    float d[32];
#pragma unroll
    for (int i = 0; i < 32; ++i) d[i] = np[i] - c[i];
#pragma unroll
    for (int o = 0; o < 16; ++o) {
      const float* wr = &w2s[(obase + o) * 64];
      float hh = 0.f;
#pragma unroll
      for (int i = 0; i < 32; ++i) hh += wr[i] * c[i] + wr[32 + i] * d[i];
      sum[o] += hh; sq[o] += hh * hh;
    }
  }
#pragma unroll
  for (int o = 0; o < 16; ++o) { stage[t * 32 + o] = sum[o]; stage[t * 32 + 16 + o] = sq[o]; }
  __syncthreads();
  if (t < 128) {
    int ch = t & 63, s = t >> 6;
    int oc2 = ch >> 4, ol = ch & 15;
    float acc = 0.f;
    for (int r = 0; r < 64; ++r) acc += stage[(r * 4 + oc2) * 32 + s * 16 + ol];
    partial[(size_t)blockIdx.x * 128 + t] = acc;
  }
}

__global__ void ec2_final_kernel(const float* __restrict__ pts2, const int* __restrict__ idx2,
                                 const float* __restrict__ w2, const float* __restrict__ stats,
                                 const float* __restrict__ g2, const float* __restrict__ b2,
                                 float* __restrict__ ec2out) {
  __shared__ float w2s[64 * 64];
  __shared__ float st[128];
  __shared__ float gb[128];
  int t = threadIdx.x;
  for (int i = t; i < 4096; i += 256) w2s[i] = w2[i];
  if (t < 128) st[t] = stats[t];
  if (t < 64) { gb[t] = g2[t]; gb[64 + t] = b2[t]; }
  __syncthreads();
  int rl = t >> 2, oc = t & 3;
  size_t g = (size_t)blockIdx.x * 64 + rl;
  size_t b = g / NN, n = g % NN;
  float c[32];
  const float* cp = pts2 + g * 32;
#pragma unroll
  for (int i = 0; i < 32; ++i) c[i] = cp[i];
  float mv[16];
#pragma unroll
  for (int o = 0; o < 16; ++o) mv[o] = -3.4e38f;
  const int* ip = idx2 + g * KK;
  int obase = oc * 16;
  for (int k = 0; k < KK; ++k) {
    int j = ip[k];
    const float* np = pts2 + (b * NN + j) * 32;
    float d[32];
#pragma unroll
    for (int i = 0; i < 32; ++i) d[i] = np[i] - c[i];
#pragma unroll
    for (int o = 0; o < 16; ++o) {
      const float* wr = &w2s[(obase + o) * 64];
      float hh = 0.f;
#pragma unroll
      for (int i = 0; i < 32; ++i) hh += wr[i] * c[i] + wr[32 + i] * d[i];
      int och = obase + o;
      float v = leakyf((hh - st[och * 2]) * st[och * 2 + 1] * gb[och] + gb[64 + och]);
      mv[o] = fmaxf(mv[o], v);
    }
  }
#pragma unroll
  for (int o = 0; o < 16; ++o) ec2out[(b * 64 + obase + o) * NN + n] = mv[o];
}

// ---------------------------------------------------------------------------
// Shortcut: sc_raw[b,o,n] = ws @ feat1
// ---------------------------------------------------------------------------
__global__ void sc_compute_kernel(const float* __restrict__ f1nc, const float* __restrict__ ws,
                                  float* __restrict__ scraw) {
  __shared__ float wss[64 * 32];
  int t = threadIdx.x;
  for (int i = t; i < 2048; i += 256) wss[i] = ws[i];
  __syncthreads();
  size_t g = (size_t)blockIdx.x * 256 + t;
  size_t b = g / NN, n = g % NN;
  float f[32];
  const float* fp = f1nc + g * 32;
#pragma unroll
  for (int i = 0; i < 32; ++i) f[i] = fp[i];
  for (int o = 0; o < 64; ++o) {
    float a = 0.f;
#pragma unroll
    for (int i = 0; i < 32; ++i) a += wss[o * 32 + i] * f[i];
    scraw[(b * 64 + o) * NN + n] = a;
  }
}

// stats over (b,n) for a (B,C,N) tensor; grid (C, nchunks)
__global__ void stats_bcn_kernel(const float* __restrict__ x, float* __restrict__ partial, int C,
                                 int nchunks) {
  __shared__ float s1[256], s2[256];
  int c = blockIdx.x, chunk = blockIdx.y;
  int t = threadIdx.x;
  int per = (BB * NN) / nchunks;
  float a = 0.f, a2 = 0.f;
  for (int s = chunk * per + t; s < (chunk + 1) * per; s += 256) {
    int b = s / NN, n = s % NN;
    float v = x[((size_t)b * C + c) * NN + n];
    a += v; a2 += v * v;
  }
  s1[t] = a; s2[t] = a2;
  __syncthreads();
  for (int off = 128; off > 0; off >>= 1) {
    if (t < off) { s1[t] += s1[t + off]; s2[t] += s2[t + off]; }
    __syncthreads();
  }
  if (t == 0) {
    partial[((size_t)c * nchunks + chunk) * 2] = s1[0];
    partial[((size_t)c * nchunks + chunk) * 2 + 1] = s2[0];
  }
}

__global__ void stats_bcn_final_kernel(const float* __restrict__ partial,
                                       float* __restrict__ stats, int C, int nchunks,
                                       float inv_cnt) {
  int c = threadIdx.x + blockIdx.x * blockDim.x;
  if (c >= C) return;
  float a = 0.f, a2 = 0.f;
  for (int k = 0; k < nchunks; ++k) {
    a += partial[((size_t)c * nchunks + k) * 2];
    a2 += partial[((size_t)c * nchunks + k) * 2 + 1];
  }
  float mean = a * inv_cnt;
  float var = a2 * inv_cnt - mean * mean;
  stats[c * 2] = mean;
  stats[c * 2 + 1] = rsqrtf(var + EPSF);
}

__global__ void fuse_feat2_kernel(float* __restrict__ ec2out, const float* __restrict__ scraw,
                                  const float* __restrict__ stats, const float* __restrict__ gs,
                                  const float* __restrict__ bs) {
  size_t i = (size_t)blockIdx.x * 256 + threadIdx.x;
  int o = (int)((i / NN) % 64);
  float v = scraw[i];
  ec2out[i] += (v - stats[o * 2]) * stats[o * 2 + 1] * gs[o] + bs[o];
}

// ---------------------------------------------------------------------------
// post_pre = wp @ concat(feat1, feat2)
// ---------------------------------------------------------------------------
__global__ void post_compute_kernel(const float* __restrict__ f1cn, const float* __restrict__ feat2,
                                    const float* __restrict__ wp, float* __restrict__ postpre) {
  __shared__ float wps[96 * 96];
  int t = threadIdx.x;
  for (int i = t; i < 9216; i += 256) wps[i] = wp[i];
  __syncthreads();
  size_t g = (size_t)blockIdx.x * 256 + t;
  size_t b = g / NN, n = g % NN;
  float msv[96];
#pragma unroll
  for (int c2 = 0; c2 < 32; ++c2) msv[c2] = f1cn[(b * 32 + c2) * NN + n];
#pragma unroll
  for (int c2 = 0; c2 < 64; ++c2) msv[32 + c2] = feat2[(b * 64 + c2) * NN + n];
  for (int o = 0; o < 96; ++o) {
    float a = 0.f;
#pragma unroll
    for (int c2 = 0; c2 < 96; ++c2) a += wps[o * 96 + c2] * msv[c2];
    postpre[(b * 96 + o) * NN + n] = a;
  }
}

__global__ void post_final_kernel(const float* __restrict__ postpre, const float* __restrict__ stats,
                                  const float* __restrict__ gp, const float* __restrict__ bp,
                                  float* __restrict__ post) {
  size_t i = (size_t)blockIdx.x * 256 + threadIdx.x;
  int o = (int)((i / NN) % 96);
  post[i] = leakyf((postpre[i] - stats[o * 2]) * stats[o * 2 + 1] * gp[o] + bp[o]);
}

__global__ void gf_reduce_kernel(const float* __restrict__ post, float* __restrict__ gf) {
  __shared__ float sm[256], ss[256];
  int b = blockIdx.x / 96, o = blockIdx.x % 96, t = threadIdx.x;
  const float* p = post + ((size_t)b * 96 + o) * NN;
  float m = -3.4e38f, s = 0.f;
  for (int n = t; n < NN; n += 256) {
    float v = p[n];
    m = fmaxf(m, v);
    s += v;
  }
  sm[t] = m; ss[t] = s;
  __syncthreads();
  for (int off = 128; off > 0; off >>= 1) {
    if (t < off) { sm[t] = fmaxf(sm[t], sm[t + off]); ss[t] += ss[t + off]; }
    __syncthreads();
  }
  if (t == 0) {
    gf[b * 198 + o] = sm[0];
    gf[b * 198 + 96 + o] = ss[0] / NN;
  }
}

// ---------------------------------------------------------------------------
// Dense head: bn over batch, leaky, tanh, sigmoid, counts
// ---------------------------------------------------------------------------
__global__ void dense_kernel(const float* __restrict__ gf, const float* __restrict__ wd1,
                             const float* __restrict__ bd1, const float* __restrict__ gd,
                             const float* __restrict__ bdv, const float* __restrict__ wd2,
                             const float* __restrict__ bd2, float* __restrict__ out, int o_vc,
                             int o_lg, int o_pb, int o_num) {
  __shared__ float hbuf[8 * 128];
  __shared__ int cnts[8];
  int t = threadIdx.x;
  if (t < 8) cnts[t] = 0;
  for (int it = 0; it < 4; ++it) {
    int i = it * 256 + t;
    int b = i >> 7, j = i & 127;
    float a = bd1[j];
    for (int c = 0; c < 198; ++c) a += gf[b * 198 + c] * wd1[j * 198 + c];
    hbuf[i] = a;
  }
  __syncthreads();
  if (t < 128) {
    float m = 0.f;
    for (int b = 0; b < 8; ++b) m += hbuf[b * 128 + t];
    m *= 0.125f;
    float v = 0.f;
    for (int b = 0; b < 8; ++b) { float d = hbuf[b * 128 + t] - m; v += d * d; }
    v *= 0.125f;
    float inv = rsqrtf(v + EPSF);
    for (int b = 0; b < 8; ++b)
      hbuf[b * 128 + t] = leakyf((hbuf[b * 128 + t] - m) * inv * gd[t] + bdv[t]);
  }
  __syncthreads();
  for (int it = 0; it < 8; ++it) {
    int i = it * 256 + t;
    int b = i >> 8, v = i & 255;
    float a = bd2[v];
    for (int j = 0; j < 128; ++j) a += hbuf[b * 128 + j] * wd2[v * 128 + j];
    float tv = tanhf(a);
    int vi = v >> 2, cch = v & 3;
    if (cch < 3) {
      out[o_vc + ((size_t)b * MAXVV + vi) * 3 + cch] = tv;
    } else {
      out[o_lg + b * MAXVV + vi] = tv;
      float pr = 1.f / (1.f + expf(-tv));
      out[o_pb + b * MAXVV + vi] = pr;
      if (pr > 0.5f) atomicAdd(&cnts[b], 1);
    }
  }
  __syncthreads();
  if (t < 8) {
    int nm = cnts[t];
    if (nm < 1) nm = 1;
    if (nm > MAXVV) nm = MAXVV;
    out[o_num + t] = (float)nm;
  }
}

// ---------------------------------------------------------------------------
// Host launch
// ---------------------------------------------------------------------------
extern "C" void kernel_launch(void* const* d_in, const int* in_sizes, int n_in, void* d_out,
                              int out_size, void* d_ws, size_t ws_size, hipStream_t stream) {
  (void)in_sizes; (void)n_in; (void)out_size; (void)ws_size;
  const float* x = (const float*)d_in[0];
  const float* w1 = (const float*)d_in[1];
  const float* g1 = (const float*)d_in[2];
  const float* b1 = (const float*)d_in[3];
  const float* w2 = (const float*)d_in[4];
  const float* g2 = (const float*)d_in[5];
  const float* b2 = (const float*)d_in[6];
  const float* ws = (const float*)d_in[7];
  const float* gs = (const float*)d_in[8];
  const float* bs = (const float*)d_in[9];
  const float* wp = (const float*)d_in[10];
  const float* gp = (const float*)d_in[11];
  const float* bp = (const float*)d_in[12];
  const float* wd1 = (const float*)d_in[13];
  const float* bd1 = (const float*)d_in[14];
  const float* gd = (const float*)d_in[15];
  const float* bd = (const float*)d_in[16];
  const float* wd2 = (const float*)d_in[17];
  const float* bd2 = (const float*)d_in[18];
  float* out = (float*)d_out;
  float* W = (float*)d_ws;

  // d_out offsets (flat, reference return order)
  const int O_VC = 0;
  const int O_LG = 1536;
  const int O_PB = 2048;
  const int O_NUM = 2560;
  const int O_GF = 2568;
  const int O_POST = 4152;
  const int O_PCA = 3149880;
  const int O_ROT = 3149928;
  const int O_CENT = 3150000;
  const int O_ALIGNED = 3150024;

  // workspace offsets (floats)
  const size_t PTS1 = 0;          // (B,N,8) padded points
  const size_t XX1 = 262144;      // (B,N)
  const size_t IDX1 = 294912;     // (B,N,20) int
  const size_t PTS2 = 950272;     // feat1 (B,N,32)
  const size_t XX2 = 1998848;     // (B,N)
  const size_t IDX2 = 2031616;    // (B,N,20) int
  const size_t F1CN = 2686976;    // feat1 (B,32,N)
  const size_t EC2OUT = 3735552;  // (B,64,N), becomes feat2 in place
  const size_t SCRAW = 5832704;   // (B,64,N)
  const size_t POSTPRE = 7929856; // (B,96,N)
  const size_t PART = 11075584;   // block partials (max 512*128)
  const size_t STATS = 11141120;  // mean/inv_std pairs

  pca_kernel<<<BB, 256, 0, stream>>>(x, out, W + PTS1, W + XX1, O_ALIGNED, O_ROT, O_CENT, O_PCA,
                                     O_GF);
  knn_wmma_kernel<8><<<128, 256, 0, stream>>>(W + PTS1, W + XX1, (int*)(W + IDX1));
  ec1_stats_kernel<<<128, 256, 0, stream>>>(W + PTS1, (const int*)(W + IDX1), w1, W + PART);
  bn_finalize_kernel<<<1, 256, 0, stream>>>(W + PART, W + STATS, 32, 128,
                                            1.f / (float)(BB * NN * KK));
  ec1_final_kernel<<<128, 256, 0, stream>>>(W + PTS1, (const int*)(W + IDX1), w1, W + STATS, g1,
                                            b1, W + F1CN, W + PTS2, W + XX2);
  knn_wmma_kernel<32><<<128, 256, 0, stream>>>(W + PTS2, W + XX2, (int*)(W + IDX2));
  ec2_stats_kernel<<<512, 256, 0, stream>>>(W + PTS2, (const int*)(W + IDX2), w2, W + PART);
  bn_finalize_kernel<<<1, 256, 0, stream>>>(W + PART, W + STATS, 64, 512,
                                            1.f / (float)(BB * NN * KK));
  ec2_final_kernel<<<512, 256, 0, stream>>>(W + PTS2, (const int*)(W + IDX2), w2, W + STATS, g2,
                                            b2, W + EC2OUT);
  sc_compute_kernel<<<128, 256, 0, stream>>>(W + PTS2, ws, W + SCRAW);
  stats_bcn_kernel<<<dim3(64, 16), 256, 0, stream>>>(W + SCRAW, W + PART, 64, 16);
  stats_bcn_final_kernel<<<1, 64, 0, stream>>>(W + PART, W + STATS, 64, 16,
                                               1.f / (float)(BB * NN));
  fuse_feat2_kernel<<<(BB * 64 * NN) / 256, 256, 0, stream>>>(W + EC2OUT, W + SCRAW, W + STATS,
                                                              gs, bs);
  post_compute_kernel<<<128, 256, 0, stream>>>(W + F1CN, W + EC2OUT, wp, W + POSTPRE);
  stats_bcn_kernel<<<dim3(96, 16), 256, 0, stream>>>(W + POSTPRE, W + PART, 96, 16);
  stats_bcn_final_kernel<<<1, 96, 0, stream>>>(W + PART, W + STATS, 96, 16,
                                               1.f / (float)(BB * NN));
  post_final_kernel<<<(BB * 96 * NN) / 256, 256, 0, stream>>>(W + POSTPRE, W + STATS, gp, bp,
                                                              out + O_POST);
  gf_reduce_kernel<<<BB * 96, 256, 0, stream>>>(out + O_POST, out + O_GF);
  dense_kernel<<<1, 256, 0, stream>>>(out + O_GF, wd1, bd1, gd, bd, wd2, bd2, out, O_VC, O_LG,
                                      O_PB, O_NUM);
}